// FeatureVolume_68762426409246
// MI455X (gfx1250) — compile-verified
//
#include <hip/hip_runtime.h>
#include <cstdint>
#include <cstddef>

// FeatureVolume cost-volume builder for MI455X (gfx1250).
//
// left/right: [B=2, H=256, W=256, C=16] f32
// out:        [B, 2D=48, H, W, 2C=32] f32   (~805 MB, pure streaming store)
//
// Strategy: 1 workgroup per (b,h) row. Stage both 16KB input rows into LDS via
// gfx1250 async global->LDS copies (ASYNCcnt path), then emit all 48 disparity
// slices from LDS with 128-bit nontemporal stores.

typedef float v4f __attribute__((ext_vector_type(4)));

#define B_      2
#define H_      256
#define W_      256
#define C_      16
#define MAXD    24
#define D2      (2 * MAXD)          // 48
#define C2      (2 * C_)            // 32
#define ROWF    (W_ * C_)           // 4096 floats per input row (16 KB)
#define NCHUNK  (ROWF / 4)          // 1024 float4 chunks per input row
#define TPB     256
#define OUTCHUNKS (D2 * W_ * (C2 / 4))          // 98304 float4 stores per row-block
#define OUT_DSTRIDE ((size_t)H_ * W_ * C2)      // 2097152 floats per disparity slice

__global__ __launch_bounds__(TPB)
void featvol_kernel(const float* __restrict__ left,
                    const float* __restrict__ right,
                    float* __restrict__ out) {
  __shared__ __align__(16) float lds[2 * ROWF];   // 32 KB: [left row | right row]

  const int tid = threadIdx.x;
  const int bh  = blockIdx.x;          // 0 .. B*H-1
  const int b   = bh >> 8;             // /256
  const int h   = bh & (H_ - 1);

  const float* lrow = left  + (size_t)bh * ROWF;
  const float* rrow = right + (size_t)bh * ROWF;

  // Low 32 bits of a generic LDS pointer == wave-relative LDS byte offset.
  const unsigned ldsL = (unsigned)(uintptr_t)&lds[0];
  const unsigned ldsR = ldsL + (unsigned)(ROWF * sizeof(float));

  // ---- Stage both rows into LDS with gfx1250 async copies (ASYNCcnt) ----
  for (int k = tid; k < NCHUNK; k += TPB) {
    const unsigned lofs = ldsL + (unsigned)k * 16u;
    const float*   ga   = lrow + k * 4;
    asm volatile("global_load_async_to_lds_b128 %0, %1, off"
                 :: "v"(lofs), "v"(ga) : "memory");
    const unsigned rofs = ldsR + (unsigned)k * 16u;
    const float*   gb   = rrow + k * 4;
    asm volatile("global_load_async_to_lds_b128 %0, %1, off"
                 :: "v"(rofs), "v"(gb) : "memory");
  }
  // Drain this wave's async copies, then sync the workgroup so every wave
  // sees both complete rows in LDS.
  asm volatile("s_wait_asynccnt 0" ::: "memory");
  __syncthreads();

  const v4f* l4 = (const v4f*)&lds[0];      // [w*4 + c4]
  const v4f* r4 = (const v4f*)&lds[ROWF];   // [w*4 + c4]

  // out[b, d, h, w, c] flat = b*D2*H*W*C2 + d*H*W*C2 + h*W*C2 + w*C2 + c
  const size_t base = (size_t)b * (D2 * OUT_DSTRIDE) + (size_t)h * (W_ * C2);

  // idx -> (d, w, c4): c4 fastest => consecutive lanes write consecutive 16B
  // (fully coalesced 4KB bursts per 256-thread iteration).
  for (int idx = tid; idx < OUTCHUNKS; idx += TPB) {
    const int c4 = idx & 7;                 // which float4 of the 32 channels
    const int w  = (idx >> 3) & (W_ - 1);
    const int d  = idx >> 11;               // disparity slot 0..47
    // disps[d] = d - MAXD; source col in right = w - disps[d] = w + MAXD - d
    const int src = w + MAXD - d;

    v4f val = {0.0f, 0.0f, 0.0f, 0.0f};     // invalid => both halves zero
    if ((unsigned)src < (unsigned)W_) {
      val = (c4 < 4) ? l4[(w << 2) + c4]            // left half: left[w]
                     : r4[(src << 2) + (c4 - 4)];   // right half: right[src]
    }

    const size_t o = base + (size_t)d * OUT_DSTRIDE
                          + (size_t)w * C2 + (size_t)(c4 << 2);
    // Streaming 805MB output: NT hint keeps the 192MB L2 clean.
    __builtin_nontemporal_store(val, (v4f*)(out + o));
  }
}

extern "C" void kernel_launch(void* const* d_in, const int* in_sizes, int n_in,
                              void* d_out, int out_size, void* d_ws, size_t ws_size,
                              hipStream_t stream) {
  const float* left  = (const float*)d_in[0];
  const float* right = (const float*)d_in[1];
  float* out = (float*)d_out;

  dim3 grid(B_ * H_);   // 512 row-blocks
  dim3 block(TPB);      // 8 wave32s
  featvol_kernel<<<grid, block, 0, stream>>>(left, right, out);
}